// HardwareRendering_57071525429737
// MI455X (gfx1250) — compile-verified
//
#include <hip/hip_runtime.h>

// ---------------------------------------------------------------------------
// Gaussian-surfel ray tracer for MI455X (gfx1250, wave32).
//  Kernel 1: per-surfel preprocess (quat->frame, precomputed dots) -> ws SoA.
//  Kernel 2: per-16-pixel tile render:
//     phase 1: TDM (tensor_load_to_lds) double-buffers a 16-surfel x 15-row
//              SoA tile into LDS; WMMA f32 16x16x4 computes all 6 ray/surfel
//              dot products per 16x16 tile; VALU computes t,u,v,alpha with a
//              fast v_rcp; valid hits pushed to per-pixel LDS lists (ds
//              atomics, key = t|alpha(f16)|id packed u64).
//     phase 2: bitonic sort of each pixel list in LDS (depth order).
//     phase 3: front-to-back compositing via wave32 prefix-product scans.
// ---------------------------------------------------------------------------

typedef __attribute__((ext_vector_type(2))) float v2f;
typedef __attribute__((ext_vector_type(8))) float v8f;
typedef __attribute__((ext_vector_type(4))) unsigned int u32x4;
typedef __attribute__((ext_vector_type(8))) int i32x8;
typedef __attribute__((ext_vector_type(4))) int i32x4;

#define MAXH 256           // per-pixel hit cap (power of two for bitonic)
#define TILE_PIX 16        // pixels per workgroup (WMMA M)
#define WAVES 8            // waves per workgroup
#define NROWS 15           // SoA rows staged per tile (n,tu,tv,mn,mu,mv,isu,isv,op)

// ---------------- surfel preprocess -> SoA in workspace --------------------
// ws layout (arrays of P floats): 0:nx 1:ny 2:nz 3:tux 4:tuy 5:tuz
// 6:tvx 7:tvy 8:tvz 9:mn 10:mu 11:mv 12:isu 13:isv 14:op 15:cr 16:cg 17:cb
__global__ void surfel_prep(const float* __restrict__ means,
                            const float* __restrict__ scales,
                            const float* __restrict__ rots,
                            const float* __restrict__ opac,
                            const float* __restrict__ cols,
                            float* __restrict__ W, int P)
{
    int p = blockIdx.x * blockDim.x + threadIdx.x;
    if (p >= P) return;
    float qw = rots[4 * p + 0], qx = rots[4 * p + 1];
    float qy = rots[4 * p + 2], qz = rots[4 * p + 3];
    float nq = sqrtf(qw * qw + qx * qx + qy * qy + qz * qz);
    float inv = 1.0f / (nq + 1e-12f);
    qw *= inv; qx *= inv; qy *= inv; qz *= inv;

    float tux = 1.0f - 2.0f * (qy * qy + qz * qz);
    float tuy = 2.0f * (qx * qy + qw * qz);
    float tuz = 2.0f * (qx * qz - qw * qy);
    float tvx = 2.0f * (qx * qy - qw * qz);
    float tvy = 1.0f - 2.0f * (qx * qx + qz * qz);
    float tvz = 2.0f * (qy * qz + qw * qx);
    float nx  = 2.0f * (qx * qz + qw * qy);
    float ny  = 2.0f * (qy * qz - qw * qx);
    float nz  = 1.0f - 2.0f * (qx * qx + qy * qy);

    float mx = means[3 * p + 0], my = means[3 * p + 1], mz = means[3 * p + 2];

    W[0 * P + p] = nx;  W[1 * P + p] = ny;  W[2 * P + p] = nz;
    W[3 * P + p] = tux; W[4 * P + p] = tuy; W[5 * P + p] = tuz;
    W[6 * P + p] = tvx; W[7 * P + p] = tvy; W[8 * P + p] = tvz;
    W[9 * P + p]  = mx * nx  + my * ny  + mz * nz;
    W[10 * P + p] = mx * tux + my * tuy + mz * tuz;
    W[11 * P + p] = mx * tvx + my * tvy + mz * tvz;
    W[12 * P + p] = 1.0f / scales[2 * p + 0];
    W[13 * P + p] = 1.0f / scales[2 * p + 1];
    W[14 * P + p] = opac[p];
    W[15 * P + p] = cols[3 * p + 0];
    W[16 * P + p] = cols[3 * p + 1];
    W[17 * P + p] = cols[3 * p + 2];
}

__device__ __forceinline__ v8f wmma_f32_4(v2f a, v2f b, v8f c) {
    return __builtin_amdgcn_wmma_f32_16x16x4_f32(false, a, false, b,
                                                 (short)0, c, false, false);
}

// Issue one TDM load: 2D tile [NROWS rows x 16 surfels] of the SoA tensor
// (row stride = P elements) into LDS at lds_off.  D# per CDNA5 ISA §8.3/8.4.
// This toolchain's builtin takes 6 args (g0, g1, g2, g3, g4, cpol).
__device__ __forceinline__ void tdm_stage_tile(const float* Wbase, int gb,
                                               unsigned lds_off, int P)
{
    unsigned long long ga = (unsigned long long)(uintptr_t)(Wbase + gb);

    u32x4 g0;
    g0[0] = 1u;                                   // count=1, no gather/restore
    g0[1] = lds_off;                              // LDS byte address
    g0[2] = (unsigned)ga;                         // global_addr[31:0]
    g0[3] = (unsigned)((ga >> 32) & 0x1FFFFFFu)   // global_addr[56:32]
          | 0x80000000u;                          // type=2 ("image")

    i32x8 g1;
    g1[0] = (int)(2u << 16);                      // data_size = 4 bytes
    g1[1] = (int)((unsigned)(P & 0xFFFF) << 16);  // tensor_dim0[15:0]
    g1[2] = (int)(((unsigned)P >> 16) & 0xFFFFu)  // tensor_dim0[31:16]
          | (18 << 16);                           // tensor_dim1 = 18 rows
    g1[3] = (16 << 16);                           // tile_dim0 = 16 surfels
    g1[4] = NROWS;                                // tile_dim1 = 15 rows
    g1[5] = P;                                    // tensor_dim0_stride
    g1[6] = 0;
    g1[7] = 0;

    i32x4 gz4 = {0, 0, 0, 0};                     // groups 2/3: NULL (<=2D)
    i32x8 gz8 = {0, 0, 0, 0, 0, 0, 0, 0};
    __builtin_amdgcn_tensor_load_to_lds(g0, g1, gz4, gz4, gz8, 0);
}

// ------------------------------- render ------------------------------------
__global__ __launch_bounds__(256)
void surfel_render(const float* __restrict__ ro_g,
                   const float* __restrict__ rd_g,
                   const float* __restrict__ bg,
                   const float* __restrict__ W,
                   float* __restrict__ out, int NPIX, int P)
{
    __shared__ unsigned long long keys[TILE_PIX][MAXH];    // 32 KB
    __shared__ float stage[WAVES][2][NROWS * 16];          // 15.4 KB (TDM dst)
    __shared__ int cnt[TILE_PIX];

    const int tid  = threadIdx.x;
    const int wave = tid >> 5;
    const int lane = tid & 31;
    const int PB   = blockIdx.x * TILE_PIX;

    if (tid < TILE_PIX) cnt[tid] = 0;
    __syncthreads();

    // ---------------- phase 1: TDM staging + WMMA hit generation -----------
    const int mrow = lane & 15;        // pixel row within tile / surfel column
    const int half = lane >> 4;        // 0: lanes 0-15, 1: lanes 16-31
    const int pix  = PB + mrow;

    const float rdx = rd_g[3 * pix + 0], rdy = rd_g[3 * pix + 1], rdz = rd_g[3 * pix + 2];
    const float rox = ro_g[3 * pix + 0], roy = ro_g[3 * pix + 1], roz = ro_g[3 * pix + 2];

    // A 16x4 f32 fragment: lanes 0-15 hold K=0,1; lanes 16-31 hold K=2,3 (pad)
    v2f a_rd, a_ro;
    a_rd.x = half ? rdz : rdx;  a_rd.y = half ? 0.0f : rdy;
    a_ro.x = half ? roz : rox;  a_ro.y = half ? 0.0f : roy;

    const v8f zc = {0.f, 0.f, 0.f, 0.f, 0.f, 0.f, 0.f, 0.f};
    const int ntile = P >> 4;

    const unsigned lds0 = (unsigned)(unsigned long long)(uintptr_t)&stage[wave][0][0];
    const unsigned lds1 = (unsigned)(unsigned long long)(uintptr_t)&stage[wave][1][0];

    // prefetch first tile for this wave
    if (wave < ntile) tdm_stage_tile(W, wave << 4, lds0, P);

    int buf = 0;
    for (int gt = wave; gt < ntile; gt += WAVES) {
        const int gnext = gt + WAVES;
        if (gnext < ntile)   // prefetch next tile into other buffer
            tdm_stage_tile(W, gnext << 4, buf ? lds0 : lds1, P);

        // wait: current tile's DMA complete (older of <=2 outstanding)
        if (gnext < ntile) __builtin_amdgcn_s_wait_tensorcnt(1);
        else               __builtin_amdgcn_s_wait_tensorcnt(0);
        asm volatile("" ::: "memory");   // order LDS reads after the DMA

        const float* S = &stage[wave][buf][0];
        const int j = mrow;

        // B 4x16 f32 fragments from the staged tile
        v2f b_n, b_tu, b_tv;
        b_n.x  = half ? S[1 * 16 + j] : S[0 * 16 + j];
        b_n.y  = half ? 0.0f          : S[2 * 16 + j];
        b_tu.x = half ? S[4 * 16 + j] : S[3 * 16 + j];
        b_tu.y = half ? 0.0f          : S[5 * 16 + j];
        b_tv.x = half ? S[7 * 16 + j] : S[6 * 16 + j];
        b_tv.y = half ? 0.0f          : S[8 * 16 + j];

        const float mn  = S[9 * 16 + j],  mu  = S[10 * 16 + j], mv = S[11 * 16 + j];
        const float isu = S[12 * 16 + j], isv = S[13 * 16 + j], op = S[14 * 16 + j];

        const v8f cden = wmma_f32_4(a_rd, b_n,  zc);  // rd . n
        const v8f crdu = wmma_f32_4(a_rd, b_tu, zc);  // rd . tu
        const v8f crdv = wmma_f32_4(a_rd, b_tv, zc);  // rd . tv
        const v8f cron = wmma_f32_4(a_ro, b_n,  zc);  // ro . n
        const v8f crou = wmma_f32_4(a_ro, b_tu, zc);  // ro . tu
        const v8f crov = wmma_f32_4(a_ro, b_tv, zc);  // ro . tv

        const int g = (gt << 4) + mrow;

        // element (M = r + 8*half, N = mrow): pixel PB+M vs surfel g
        #pragma unroll
        for (int r = 0; r < 8; ++r) {
            const float den  = cden[r];
            const float ad   = fabsf(den);
            const float sden = (ad < 1e-8f) ? 1e-8f : den;
            const float t    = (mn - cron[r]) * __builtin_amdgcn_rcpf(sden);
            const float uu   = (crou[r] - mu + t * crdu[r]) * isu;
            const float vv   = (crov[r] - mv + t * crdv[r]) * isv;
            const float r2   = uu * uu + vv * vv;
            const float G    = __expf(-0.5f * r2);
            const float alpha = fminf(op * G, 0.99f);
            if ((t > 0.2f) && (r2 < 9.0f) && (ad > 1e-8f) && (alpha > (1.0f / 255.0f))) {
                const int m = r + (half << 3);
                const int slot = atomicAdd(&cnt[m], 1);
                if (slot < MAXH) {
                    const unsigned short ab =
                        __builtin_bit_cast(unsigned short, (_Float16)alpha);
                    keys[m][slot] = ((unsigned long long)__float_as_uint(t) << 32)
                                  | ((unsigned long long)ab << 12)
                                  | (unsigned long long)(g & 0xFFF);
                }
            }
        }
        buf ^= 1;
    }
    __syncthreads();
    if (tid < TILE_PIX) cnt[tid] = (cnt[tid] < MAXH) ? cnt[tid] : MAXH;
    __syncthreads();

    // ---------------- phases 2+3: sort + composite (wave handles 2 lists) --
    for (int li = 0; li < 2; ++li) {
        const int m = wave + (li << 3);
        const int n = cnt[m];

        for (int i = n + lane; i < MAXH; i += 32) keys[m][i] = ~0ULL;  // +inf
        __syncthreads();

        // bitonic sort ascending by packed (t,alpha,id) key
        for (unsigned k = 2; k <= MAXH; k <<= 1) {
            for (unsigned j = k >> 1; j > 0; j >>= 1) {
                for (unsigned i = lane; i < MAXH; i += 32) {
                    const unsigned ixj = i ^ j;
                    if (ixj > i) {
                        const unsigned long long va = keys[m][i];
                        const unsigned long long vb = keys[m][ixj];
                        const bool up = ((i & k) == 0);
                        if ((va > vb) == up) { keys[m][i] = vb; keys[m][ixj] = va; }
                    }
                }
                __syncthreads();
            }
        }

        // front-to-back compositing, 32 hits at a time via prefix products
        float T = 1.0f;
        float aR = 0.f, aG = 0.f, aB = 0.f, aD = 0.f, aA = 0.f;
        float aNx = 0.f, aNy = 0.f, aNz = 0.f;
        for (int base = 0; base < n; base += 32) {
            const int idx = base + lane;
            const bool act = idx < n;
            const unsigned long long key = act ? keys[m][idx] : 0ULL;
            const float t = act ? __uint_as_float((unsigned)(key >> 32)) : 0.0f;
            float alpha = 0.0f;
            if (act)
                alpha = (float)__builtin_bit_cast(
                    _Float16, (unsigned short)((key >> 12) & 0xFFFFu));
            const unsigned g = (unsigned)(key & 0xFFFu);

            float incl = 1.0f - alpha;            // inclusive scan of (1-a)
            #pragma unroll
            for (int off = 1; off < 32; off <<= 1) {
                const float vprev = __shfl_up(incl, off, 32);
                if (lane >= off) incl *= vprev;
            }
            float excl = __shfl_up(incl, 1, 32);
            if (lane == 0) excl = 1.0f;

            const float wgt = alpha * T * excl;
            aR += wgt * W[15 * P + g];
            aG += wgt * W[16 * P + g];
            aB += wgt * W[17 * P + g];
            aD += wgt * t;
            aA += wgt;
            aNx += wgt * W[0 * P + g];
            aNy += wgt * W[1 * P + g];
            aNz += wgt * W[2 * P + g];

            T *= __shfl(incl, 31, 32);
            if (T < 1e-4f) break;
        }

        #pragma unroll
        for (int off = 16; off > 0; off >>= 1) {
            aR  += __shfl_xor(aR,  off, 32);
            aG  += __shfl_xor(aG,  off, 32);
            aB  += __shfl_xor(aB,  off, 32);
            aD  += __shfl_xor(aD,  off, 32);
            aA  += __shfl_xor(aA,  off, 32);
            aNx += __shfl_xor(aNx, off, 32);
            aNy += __shfl_xor(aNy, off, 32);
            aNz += __shfl_xor(aNz, off, 32);
        }
        if (lane == 0) {
            const int p = PB + m;
            out[3 * p + 0] = aR + (1.0f - aA) * bg[0];
            out[3 * p + 1] = aG + (1.0f - aA) * bg[1];
            out[3 * p + 2] = aB + (1.0f - aA) * bg[2];
            out[3 * NPIX + p] = aD;
            out[4 * NPIX + p] = aA;
            out[5 * NPIX + 3 * p + 0] = aNx;
            out[5 * NPIX + 3 * p + 1] = aNy;
            out[5 * NPIX + 3 * p + 2] = aNz;
        }
        __syncthreads();
    }
}

// ---------------------------------------------------------------------------
extern "C" void kernel_launch(void* const* d_in, const int* in_sizes, int n_in,
                              void* d_out, int out_size, void* d_ws, size_t ws_size,
                              hipStream_t stream)
{
    const float* ro     = (const float*)d_in[0];
    const float* rd     = (const float*)d_in[1];
    const float* means  = (const float*)d_in[2];
    const float* scales = (const float*)d_in[3];
    const float* rots   = (const float*)d_in[4];
    const float* opac   = (const float*)d_in[5];
    const float* cols   = (const float*)d_in[6];
    const float* bg     = (const float*)d_in[7];

    const int NPIX = in_sizes[0] / 3;   // 16384
    const int P    = in_sizes[2] / 3;   // 4096

    float* W   = (float*)d_ws;          // 18*P floats (~295 KB)
    float* out = (float*)d_out;

    surfel_prep<<<(P + 255) / 256, 256, 0, stream>>>(means, scales, rots, opac,
                                                     cols, W, P);
    surfel_render<<<NPIX / TILE_PIX, 256, 0, stream>>>(ro, rd, bg, W, out,
                                                       NPIX, P);
}